// ActorCriticCriterionAIC_57990648430743
// MI455X (gfx1250) — compile-verified
//
#include <hip/hip_runtime.h>
#include <hip/hip_bf16.h>
#include <stdint.h>

// Problem shape (fixed by the reference)
#define BATCH 8192
#define TLEN  1024

typedef __attribute__((ext_vector_type(2))) float v2f;
typedef __attribute__((ext_vector_type(8))) float v8f;

// Low 32 bits of a flat shared-space pointer == LDS byte offset (aperture
// mapping truncates to addr[31:0]; both flat and async paths add LDS_BASE).
__device__ __forceinline__ uint32_t lds_byte_off(const void* p) {
    return (uint32_t)(uintptr_t)p;
}

// ---------------------------------------------------------------------------
// Kernel 1: one workgroup per row. Async-DMA the three float rows into LDS
// (overlapped with the EOS scan over seq), find first EOS, reduce the masked
// -logp*(reward-value) sum, and emit per-row partials (deterministic).
// ---------------------------------------------------------------------------
__global__ __launch_bounds__(256) void row_reduce_kernel(
    const int*   __restrict__ seq,
    const float* __restrict__ logp,
    const float* __restrict__ value,
    const float* __restrict__ reward,
    float* __restrict__ num_part,
    float* __restrict__ den_part) {

    __shared__ float s_lp[TLEN];
    __shared__ float s_r[TLEN];
    __shared__ float s_v[TLEN];
    __shared__ int   s_min;
    __shared__ float s_wsum[8];

    const int  tid  = threadIdx.x;       // 0..255, owns tokens [4*tid, 4*tid+4)
    const int  t0   = tid * 4;
    const long base = (long)blockIdx.x * TLEN + t0;

    // --- CDNA5 async copy: 16B per thread per array, global -> LDS (ASYNCcnt)
    {
        uint32_t dl = lds_byte_off(&s_lp[t0]);
        uint32_t dr = lds_byte_off(&s_r[t0]);
        uint32_t dv = lds_byte_off(&s_v[t0]);
        const float* gl = logp   + base;
        const float* gr = reward + base;
        const float* gv = value  + base;
        asm volatile("global_load_async_to_lds_b128 %0, %1, off"
                     :: "v"(dl), "v"(gl) : "memory");
        asm volatile("global_load_async_to_lds_b128 %0, %1, off"
                     :: "v"(dr), "v"(gr) : "memory");
        asm volatile("global_load_async_to_lds_b128 %0, %1, off"
                     :: "v"(dv), "v"(gv) : "memory");
    }

    if (tid == 0) s_min = TLEN;

    // --- EOS scan over this thread's 4 tokens (128-bit coalesced load)
    int4 sv = *(const int4*)(seq + base);
    int firstz = TLEN;
    if (sv.w == 0) firstz = t0 + 3;
    if (sv.z == 0) firstz = t0 + 2;
    if (sv.y == 0) firstz = t0 + 1;
    if (sv.x == 0) firstz = t0 + 0;

    __syncthreads();                       // s_min initialized
    if (firstz < TLEN) atomicMin(&s_min, firstz);

    // Wait for our own async-LDS copies, then make everything WG-visible.
    asm volatile("s_wait_asynccnt 0" ::: "memory");
    __syncthreads();

    // mask[t] == 1  <=>  t < min(first_eos + 1, T)
    const int limit = min(s_min + 1, TLEN);

    float acc = 0.0f;
    #pragma unroll
    for (int j = 0; j < 4; ++j) {
        int t = t0 + j;
        if (t < limit) acc += s_lp[t] * (s_r[t] - s_v[t]);
    }

    // wave32 tree reduce, then 8 wave partials via LDS (fixed order -> determ.)
    #pragma unroll
    for (int off = 16; off > 0; off >>= 1)
        acc += __shfl_down(acc, off, 32);
    if ((tid & 31) == 0) s_wsum[tid >> 5] = acc;
    __syncthreads();
    if (tid == 0) {
        float s = 0.0f;
        #pragma unroll
        for (int w = 0; w < 8; ++w) s += s_wsum[w];
        num_part[blockIdx.x] = -s;            // output = -logp*(r-v)*mask
        den_part[blockIdx.x] = (float)limit;  // exact: integer < 2^24
    }
}

// ---------------------------------------------------------------------------
// Kernel 2: single wave32. WMMA-accumulated sum of the 8192 row partials for
// numerator and denominator. B = all-ones makes V_WMMA_F32_16X16X4_F32 a
// 64-wide accumulating adder: D[m][n] = sum_k A[m][k] + C[m][n]; any bijective
// packing of 64 elements into A is valid for a full sum. Column 0 of C/D is
// lanes 0 (M=0..7) and 16 (M=8..15) across VGPRs 0..7.
// ---------------------------------------------------------------------------
__global__ __launch_bounds__(32) void final_reduce_kernel(
    const float* __restrict__ num_part,
    const float* __restrict__ den_part,
    float* __restrict__ out) {

    const int lane = threadIdx.x;  // 0..31, all lanes active (EXEC all-1s)
    v2f ones; ones.x = 1.0f; ones.y = 1.0f;
    v8f cn = {};
    v8f cd = {};

    for (int i = 0; i < BATCH; i += 64) {
        v2f a = *(const v2f*)(num_part + i + lane * 2);
        v2f b = *(const v2f*)(den_part + i + lane * 2);
        cn = __builtin_amdgcn_wmma_f32_16x16x4_f32(
                 false, a, false, ones, (short)0, cn, false, false);
        cd = __builtin_amdgcn_wmma_f32_16x16x4_f32(
                 false, b, false, ones, (short)0, cd, false, false);
    }

    float sn = cn[0] + cn[1] + cn[2] + cn[3] + cn[4] + cn[5] + cn[6] + cn[7];
    float sd = cd[0] + cd[1] + cd[2] + cd[3] + cd[4] + cd[5] + cd[6] + cd[7];
    // combine the two halves of column 0 (lane 0 <-> lane 16)
    sn += __shfl_xor(sn, 16, 32);
    sd += __shfl_xor(sd, 16, 32);

    if (lane == 0) out[0] = sn / sd;
}

// ---------------------------------------------------------------------------
extern "C" void kernel_launch(void* const* d_in, const int* in_sizes, int n_in,
                              void* d_out, int out_size, void* d_ws, size_t ws_size,
                              hipStream_t stream) {
    (void)in_sizes; (void)n_in; (void)out_size; (void)ws_size;

    const int*   seq    = (const int*)  d_in[0];  // sample_seq        [B,T] i32
    const float* logp   = (const float*)d_in[1];  // sample_seqLogprobs[B,T] f32
    const float* value  = (const float*)d_in[2];  // sample_value      [B,T] f32
    const float* reward = (const float*)d_in[3];  // sample_reward     [B,T] f32

    float* num_part = (float*)d_ws;               // [BATCH]
    float* den_part = num_part + BATCH;           // [BATCH]  (64 KB total)

    row_reduce_kernel<<<BATCH, 256, 0, stream>>>(seq, logp, value, reward,
                                                 num_part, den_part);
    final_reduce_kernel<<<1, 32, 0, stream>>>(num_part, den_part, (float*)d_out);
}